// SecurityGNNLayer_88021059764887
// MI455X (gfx1250) — compile-verified
//
#include <hip/hip_runtime.h>
#include <hip/hip_bf16.h>
#include <math.h>

#define NN  50000
#define EE  800000
#define DD  128
#define HH  8
#define DHH 16
#define SS  10

typedef float v2f __attribute__((ext_vector_type(2)));
typedef float v8f __attribute__((ext_vector_type(8)));

// float atomic max via sign-split trick (valid when slot initialized to -inf)
__device__ __forceinline__ void atomic_max_f32(float* addr, float val) {
    if (val >= 0.0f) atomicMax((int*)addr, __float_as_int(val));
    else             atomicMin((unsigned int*)addr, __float_as_uint(val));
}

// C[M,128] = act(A[M,128] @ W[128,128] + bias), M multiple of 16.
// One wave -> one 16x16 tile via v_wmma_f32_16x16x4_f32 (K-loop of 32).
// Layout per CDNA5 ISA 7.12.2:
//   A 16x4 f32 : lane l(0..15)=row M, VGPR j holds K = j + 2*half
//   B  4x16 f32: lane l(0..15)=col N, VGPR j holds K = j + 2*half
//   C/D 16x16  : VGPR r -> row M = r + 8*half, lane l -> col N
template<int RELU>
__global__ __launch_bounds__(256) void gemm128_wmma_f32(
    const float* __restrict__ A, const float* __restrict__ W,
    const float* __restrict__ bias, float* __restrict__ C)
{
    const int wave = threadIdx.x >> 5;
    const int lane = threadIdx.x & 31;
    const int half = lane >> 4;
    const int l    = lane & 15;
    const int m0   = blockIdx.x << 4;
    const int n0   = wave << 4;

    const float* __restrict__ Arow = A + (size_t)(m0 + l) * DD;
    const float* __restrict__ Wcol = W + n0 + l;

    v8f acc = {};
    #pragma unroll 4
    for (int kk = 0; kk < DD; kk += 4) {
        const int kb = kk + 2 * half;
        v2f a, b;
        a.x = Arow[kb + 0];
        a.y = Arow[kb + 1];
        b.x = Wcol[(size_t)(kb + 0) * DD];
        b.y = Wcol[(size_t)(kb + 1) * DD];
        acc = __builtin_amdgcn_wmma_f32_16x16x4_f32(
                  false, a, false, b, (short)0, acc, false, false);
    }
    const float bv = bias[n0 + l];
    #pragma unroll
    for (int r = 0; r < 8; ++r) {
        float o = acc[r] + bv;
        if (RELU) o = fmaxf(o, 0.0f);
        C[(size_t)(m0 + r + 8 * half) * DD + n0 + l] = o;
    }
}

__global__ void fill_f32(float* __restrict__ p, float v, int n) {
    int i = blockIdx.x * blockDim.x + threadIdx.x;
    if (i < n) p[i] = v;
}

// sec_bias[N,H] = security @ Wsec + bsec   (tiny K=10 GEMM, scalar)
__global__ void sec_bias_kernel(const float* __restrict__ sf,
                                const float* __restrict__ Wsec,
                                const float* __restrict__ bsec,
                                float* __restrict__ out)
{
    int i = blockIdx.x * blockDim.x + threadIdx.x;
    if (i >= NN * HH) return;
    int n = i >> 3, h = i & 7;
    float acc = bsec[h];
    #pragma unroll
    for (int s = 0; s < SS; ++s) acc = fmaf(sf[n * SS + s], Wsec[s * HH + h], acc);
    out[i] = acc;
}

// scores[e,h] = dot(q[dst,h,:], k[src,h,:]) / 4 + sec_bias[src,h]; atomic max into m[dst,h]
// Head slices are 64B-aligned -> float4 (global_load_b128) gathers.
__global__ __launch_bounds__(256) void edge_scores(
    const long long* __restrict__ ei, const float* __restrict__ q,
    const float* __restrict__ k, const float* __restrict__ secb,
    float* __restrict__ scores, float* __restrict__ mbuf)
{
    int i = blockIdx.x * blockDim.x + threadIdx.x;
    if (i >= EE * HH) return;
    int e = i >> 3, h = i & 7;
    int src = (int)ei[e];
    int dst = (int)ei[EE + e];
    const float4* qp = (const float4*)(q + (size_t)dst * DD + h * DHH);
    const float4* kp = (const float4*)(k + (size_t)src * DD + h * DHH);
    float acc = 0.0f;
    #pragma unroll
    for (int j = 0; j < DHH / 4; ++j) {
        float4 a = qp[j], b = kp[j];
        acc = fmaf(a.x, b.x, acc);
        acc = fmaf(a.y, b.y, acc);
        acc = fmaf(a.z, b.z, acc);
        acc = fmaf(a.w, b.w, acc);
    }
    float sc = acc * 0.25f + secb[src * HH + h];   // 1/sqrt(16) = 0.25
    scores[i] = sc;
    atomic_max_f32(&mbuf[dst * HH + h], sc);
}

// e = exp(score - m[dst]); atomic add into s[dst]
__global__ __launch_bounds__(256) void edge_exp(
    const long long* __restrict__ ei, float* __restrict__ scores,
    const float* __restrict__ mbuf, float* __restrict__ sbuf)
{
    int i = blockIdx.x * blockDim.x + threadIdx.x;
    if (i >= EE * HH) return;
    int e = i >> 3, h = i & 7;
    int dst = (int)ei[EE + e];
    float ev = expf(scores[i] - mbuf[dst * HH + h]);
    scores[i] = ev;
    atomicAdd(&sbuf[dst * HH + h], ev);
}

// attn = e / (s[dst]+1e-9); write attn; agg[dst,h,:] += attn * v[src,h,:]
__global__ __launch_bounds__(256) void edge_attn_agg(
    const long long* __restrict__ ei, const float* __restrict__ evals,
    const float* __restrict__ sbuf, const float* __restrict__ v,
    float* __restrict__ attn_out, float* __restrict__ agg)
{
    int i = blockIdx.x * blockDim.x + threadIdx.x;
    if (i >= EE * HH) return;
    int e = i >> 3, h = i & 7;
    int src = (int)ei[e];
    int dst = (int)ei[EE + e];
    float a = evals[i] / (sbuf[dst * HH + h] + 1e-9f);
    attn_out[i] = a;
    const float4* vp = (const float4*)(v + (size_t)src * DD + h * DHH);
    float* ap = agg + (size_t)dst * DD + h * DHH;
    #pragma unroll
    for (int j = 0; j < DHH / 4; ++j) {
        float4 vv = vp[j];
        atomicAdd(&ap[4 * j + 0], a * vv.x);
        atomicAdd(&ap[4 * j + 1], a * vv.y);
        atomicAdd(&ap[4 * j + 2], a * vv.z);
        atomicAdd(&ap[4 * j + 3], a * vv.w);
    }
}

// out[row] = LayerNorm(A[row] + B[row]) * g + beta ; one wave per 128-wide row.
// Lane-contiguous float4 accesses -> b128 loads/stores, wave32 shfl reductions.
__global__ __launch_bounds__(256) void ln_residual(
    const float* __restrict__ A, const float* __restrict__ Bm,
    const float* __restrict__ g, const float* __restrict__ beta,
    float* __restrict__ out)
{
    int row = blockIdx.x * 8 + (threadIdx.x >> 5);
    if (row >= NN) return;
    int lane = threadIdx.x & 31;
    const float4* ar = (const float4*)(A  + (size_t)row * DD);
    const float4* br = (const float4*)(Bm + (size_t)row * DD);
    float4 xa = ar[lane], xb = br[lane];
    float4 x = make_float4(xa.x + xb.x, xa.y + xb.y, xa.z + xb.z, xa.w + xb.w);
    float sum = x.x + x.y + x.z + x.w;
    #pragma unroll
    for (int o = 16; o > 0; o >>= 1) sum += __shfl_xor(sum, o, 32);
    float mean = sum * (1.0f / 128.0f);
    float dx = x.x - mean, dy = x.y - mean, dz = x.z - mean, dw = x.w - mean;
    float var = dx * dx + dy * dy + dz * dz + dw * dw;
    #pragma unroll
    for (int o = 16; o > 0; o >>= 1) var += __shfl_xor(var, o, 32);
    float inv = rsqrtf(var * (1.0f / 128.0f) + 1e-5f);
    float4 gg = ((const float4*)g)[lane];
    float4 bb = ((const float4*)beta)[lane];
    float4 o4 = make_float4(dx * inv * gg.x + bb.x,
                            dy * inv * gg.y + bb.y,
                            dz * inv * gg.z + bb.z,
                            dw * inv * gg.w + bb.w);
    ((float4*)(out + (size_t)row * DD))[lane] = o4;
}

extern "C" void kernel_launch(void* const* d_in, const int* in_sizes, int n_in,
                              void* d_out, int out_size, void* d_ws, size_t ws_size,
                              hipStream_t stream) {
    const float*     x    = (const float*)d_in[0];
    const long long* ei   = (const long long*)d_in[1];   // int64 edge_index [2,E]
    const float*     sf   = (const float*)d_in[2];
    const float*     W1   = (const float*)d_in[3];  const float* b1  = (const float*)d_in[4];
    const float*     W2   = (const float*)d_in[5];  const float* b2  = (const float*)d_in[6];
    const float*     Wq   = (const float*)d_in[7];  const float* bq  = (const float*)d_in[8];
    const float*     Wk   = (const float*)d_in[9];  const float* bk  = (const float*)d_in[10];
    const float*     Wv   = (const float*)d_in[11]; const float* bv  = (const float*)d_in[12];
    const float*     Wsec = (const float*)d_in[13]; const float* bsec= (const float*)d_in[14];
    const float*     Wo   = (const float*)d_in[15]; const float* bo  = (const float*)d_in[16];
    const float*     g1   = (const float*)d_in[17]; const float* be1 = (const float*)d_in[18];
    const float*     g2   = (const float*)d_in[19]; const float* be2 = (const float*)d_in[20];

    float* ws = (float*)d_ws;
    const size_t ND = (size_t)NN * DD;
    float* B0   = ws;                                 // h, later tmp
    float* B1   = ws + ND;                            // q, later agg
    float* B2   = ws + 2 * ND;                        // k, later attended/out1
    float* B3   = ws + 3 * ND;                        // v
    float* B4   = ws + 4 * ND;                        // scores / e  [E,H]
    float* Mb   = B4 + (size_t)EE * HH;               // segment max [N,H]
    float* Sb   = Mb + (size_t)NN * HH;               // segment sum [N,H]
    float* SecB = Sb + (size_t)NN * HH;               // sec bias    [N,H]

    float* out      = (float*)d_out;                  // [N,128]
    float* attn_out = out + ND;                       // [E,8]

    const int GEMM_GRID = NN / 16;                    // 3125, exact
    dim3 blk(256);
    const int nh = NN * HH, eh = EE * HH, nd = (int)ND;

    // h = relu(x @ W1 + b1); q,k,v projections
    gemm128_wmma_f32<1><<<GEMM_GRID, blk, 0, stream>>>(x,  W1, b1, B0);
    gemm128_wmma_f32<0><<<GEMM_GRID, blk, 0, stream>>>(B0, Wq, bq, B1);
    gemm128_wmma_f32<0><<<GEMM_GRID, blk, 0, stream>>>(B0, Wk, bk, B2);
    gemm128_wmma_f32<0><<<GEMM_GRID, blk, 0, stream>>>(B0, Wv, bv, B3);

    sec_bias_kernel<<<(nh + 255) / 256, blk, 0, stream>>>(sf, Wsec, bsec, SecB);
    fill_f32<<<(nh + 255) / 256, blk, 0, stream>>>(Mb, -INFINITY, nh);
    fill_f32<<<(nh + 255) / 256, blk, 0, stream>>>(Sb, 0.0f, nh);

    edge_scores<<<(eh + 255) / 256, blk, 0, stream>>>(ei, B1, B2, SecB, B4, Mb);
    edge_exp   <<<(eh + 255) / 256, blk, 0, stream>>>(ei, B4, Mb, Sb);
    fill_f32   <<<(nd + 255) / 256, blk, 0, stream>>>(B1, 0.0f, nd);          // agg = 0 (reuse q)
    edge_attn_agg<<<(eh + 255) / 256, blk, 0, stream>>>(ei, B4, Sb, B3, attn_out, B1);

    // attended = agg @ Wo + bo (into k's slot); out1 = LN(h + attended) in place
    gemm128_wmma_f32<0><<<GEMM_GRID, blk, 0, stream>>>(B1, Wo, bo, B2);
    ln_residual<<<(NN + 7) / 8, blk, 0, stream>>>(B0, B2, g1, be1, B2);
    // tmp = relu(out1 @ W2 + b2) (into h's slot); out = LN(out1 + tmp)
    gemm128_wmma_f32<1><<<GEMM_GRID, blk, 0, stream>>>(B2, W2, b2, B0);
    ln_residual<<<(NN + 7) / 8, blk, 0, stream>>>(B2, B0, g2, be2, out);
}